// UltimatePrecisionNKATHamiltonian_16441134809569
// MI455X (gfx1250) — compile-verified
//
#include <hip/hip_runtime.h>

#define DIM   540
#define BATCH 64

// =====================================================================
// Kernel 1: bulk zero-fill of the [B,540,540] complex128 output using
// CDNA5 async LDS->global stores (ASYNCcnt path).
//
// Each wave owns a contiguous 2 KB chunk per iteration and issues four
// b128 async stores off one address computation using offset: immediates
// (0/512/1024/1536). NOTE: the ISA adds INST_OFFSET to BOTH the global
// address and the LDS source address, so the LDS zero buffer is 2 KB
// (lane*16 + 1536 + 16 == 2048 exactly). Stores are marked non-temporal:
// the 298.6 MB stream exceeds the 192 MB L2 and is never re-read on-device.
// =====================================================================
__global__ void nkat_zero_fill_async(double* __restrict__ out, size_t total16)
{
  __shared__ __align__(16) double zbuf[256];        // 2 KB of zeros in LDS
  if (threadIdx.x < 256) zbuf[threadIdx.x & 255] = 0.0;
  __syncthreads();                                   // DS writes visible before DMA reads

  const unsigned lane   = threadIdx.x & 31u;
  const unsigned ldsoff = (unsigned)(unsigned long long)(&zbuf[0]) + lane * 16u;

  const unsigned long long base = (unsigned long long)out;

  // one 2 KB chunk (128 x 16 B units) per wave per iteration
  size_t wave   = ((size_t)blockIdx.x * blockDim.x + threadIdx.x) >> 5;
  size_t nwaves = ((size_t)gridDim.x * blockDim.x) >> 5;
  size_t nchunk = total16 >> 7;                      // 18,662,400 / 128 = 145,800 exact

  for (size_t c = wave; c < nchunk; c += nwaves) {
    unsigned long long addr = base + (c << 11) + (unsigned long long)(lane << 4);
    asm volatile(
      "global_store_async_from_lds_b128 %0, %1, off th:TH_STORE_NT\n\t"
      "global_store_async_from_lds_b128 %0, %1, off offset:512 th:TH_STORE_NT\n\t"
      "global_store_async_from_lds_b128 %0, %1, off offset:1024 th:TH_STORE_NT\n\t"
      "global_store_async_from_lds_b128 %0, %1, off offset:1536 th:TH_STORE_NT"
      :: "v"(addr), "v"(ldsoff) : "memory");
  }
  // drain ASYNCcnt before the wave (and its LDS allocation) retires
  asm volatile("s_wait_asynccnt 0" ::: "memory");
}

// =====================================================================
// Kernel 2: exact fp64 band (|i-j| <= 2) — diagonal zeta term with all
// reference guards + prime / kappa / quantum corrections + REG.
// =====================================================================
__device__ __constant__ int NC_PRIMES[30] = {
    2,3,5,7,11,13,17,19,23,29,31,37,41,43,47,53,59,61,
    67,71,73,79,83,89,97,101,103,107,109,113};

__device__ __forceinline__ bool is_nc_prime(int p)
{
  bool r = false;
  #pragma unroll
  for (int k = 0; k < 30; ++k) r |= (p == NC_PRIMES[k]);
  return r;
}

// kappa base coefficient kc[k], k in [0,40)
__device__ __forceinline__ double kappa_kc(int k)
{
  const double KAPPA_A = 1e-16 * 25.0;              // 2.5e-15
  double kn   = (double)(k + 1);
  double mink = 1.0 / sqrt(1.0 + (kn * KAPPA_A) * (kn * KAPPA_A));
  return KAPPA_A * kn * log(kn + 1.0) * mink * 1e-8;
}

__device__ __forceinline__ void store_c(double* __restrict__ out,
                                        int b, int i, int j,
                                        double re, double im)
{
  size_t off = ((size_t)(b * DIM + i) * DIM + j) * 2;
  double2 v; v.x = re; v.y = im;
  *reinterpret_cast<double2*>(out + off) = v;       // 16 B store
}

__global__ void nkat_band_kernel(const float* __restrict__ s_real,
                                 const float* __restrict__ s_imag,
                                 double* __restrict__ out)
{
  const double THETA_A = 1e-24 * 50.0;              // 5e-23
  const double TOL     = 1e-18;
  const double REG     = 1e-15;

  int t = blockIdx.x * blockDim.x + threadIdx.x;
  if (t >= BATCH * DIM) return;
  int b = t / DIM;
  int i = t - b * DIM;

  double sr = (double)s_real[b];
  double si = (double)s_imag[b];

  // ---------------- diagonal: zeta = 1 / n^s with guards ----------------
  double n    = (double)(i + 1);
  double logn = log(n);                             // == log(p) for p = i+1
  double lr   = -sr * logn;
  double li   =  si * logn;
  double e    = exp(lr);
  double pr   = e * cos(li);
  double pim  = e * sin(li);
  if (lr < -100.0 || lr > 100.0) { pr = 1e-50; pim = 0.0; }
  double mag2 = pr * pr + pim * pim;
  double zr, zi;
  if (sqrt(mag2) < 1e-100) { zr = 1e-50; zi = 0.0; }
  else                     { zr = pr / mag2; zi = -pim / mag2; }
  double dr, di;
  if (sqrt(zr * zr + zi * zi) > TOL) { dr = zr;  di = zi;  }
  else                               { dr = TOL; di = 0.0; }

  const bool   pr_ip1 = is_nc_prime(i + 1);         // H[p-1, *] row (p = i+1)
  const bool   pr_i   = is_nc_prime(i);             // H[p,   *] row (p = i)
  const double conv   = 1.0 / (1.0 + fabs(si) * 0.001);
  const double mass   = 0.5 - sr;

  if (pr_ip1)  dr += THETA_A * logn * 1e-6 * 0.1;   // NC diag (real)
  if (i < 40)  dr += kappa_kc(i) * 0.01 * mass;     // kappa diag (real)
  if (i < 25)  dr += (1e-10 / (n * n)) * conv;      // quantum diag (real)
  dr += REG;                                        // stabilization
  store_c(out, b, i, i, dr, di);

  // ---------------- j = i+1 ----------------
  if (i + 1 < DIM) {
    double ore = 0.0, oim = 0.0; bool w = false;
    if (i < 40)  { ore += kappa_kc(i) * 0.005;            w = true; }
    if (pr_ip1)  { oim += THETA_A * logn * 1e-6 * 0.05;   w = true; }   // (p-1,p)
    if (w) store_c(out, b, i, i + 1, ore, oim);
  }

  // ---------------- j = i-1 ----------------
  if (i >= 1) {
    double ore = 0.0, oim = 0.0; bool w = false;
    if (i - 1 < 40) { ore += kappa_kc(i - 1) * 0.005;     w = true; }
    if (pr_i) { oim += THETA_A * log((double)i) * 1e-6 * 0.05; w = true; } // (p,p-1)
    if (w) store_c(out, b, i, i - 1, ore, oim);
  }

  // ---------------- j = i+2 (quantum, +Q_NL) ----------------
  if (i < 25) {
    double qnl = (1e-10 / (n * n)) * 0.01 / (double)(i + 3); // imag coeff
    store_c(out, b, i, i + 2, 0.0, qnl * conv);
  }

  // ---------------- j = i-2 (quantum, -Q_NL) ----------------
  if (i >= 2 && i - 2 < 25) {
    double kn  = (double)(i - 1);                    // (k+1) with k = i-2
    double qnl = (1e-10 / (kn * kn)) * 0.01 / (double)(i + 1); // (k+3) = i+1
    store_c(out, b, i, i - 2, 0.0, -qnl * conv);
  }
}

// =====================================================================
extern "C" void kernel_launch(void* const* d_in, const int* in_sizes, int n_in,
                              void* d_out, int out_size, void* d_ws, size_t ws_size,
                              hipStream_t stream)
{
  const float* s_real = (const float*)d_in[0];
  const float* s_imag = (const float*)d_in[1];
  double* out = (double*)d_out;                     // complex128 -> interleaved f64

  const size_t TOTAL16 = (size_t)BATCH * DIM * DIM; // 18,662,400 x 16 B = 298.6 MB

  // 1) stream zeros: async LDS->global b128 stores, 2 KB/wave/iter, NT hint
  nkat_zero_fill_async<<<4096, 256, 0, stream>>>(out, TOTAL16);

  // 2) overwrite the |i-j|<=2 band with exact fp64 values
  int nthreads = BATCH * DIM;                       // 34,560
  nkat_band_kernel<<<(nthreads + 255) / 256, 256, 0, stream>>>(s_real, s_imag, out);
}